// LatticeVectorQuantization_28879360098765
// MI455X (gfx1250) — compile-verified
//
#include <hip/hip_runtime.h>

typedef __attribute__((ext_vector_type(2))) float v2f;
typedef __attribute__((ext_vector_type(8))) float v8f;

// ws layout (floats): [0,64) normalized basis (8x8 row-major),
//                     [64, 64+2048) G = grid@basis (256x8 row-major),
//                     [2112, 2368) gg[j] = ||G_j||^2
__global__ void lvq_prep_kernel(const float* __restrict__ bp, float* __restrict__ ws) {
  int j = threadIdx.x;  // 0..255 == grid row index
  float det = 1.f;
#pragma unroll
  for (int i = 0; i < 8; ++i) det *= bp[i * 8 + i];
  float inv = 1.f / fabsf(det);
  float bas[8][8];
#pragma unroll
  for (int i = 0; i < 8; ++i)
#pragma unroll
    for (int k = 0; k < 8; ++k)
      bas[i][k] = (k <= i) ? bp[i * 8 + k] * inv : 0.f;
  if (j < 64) ws[j] = bas[j >> 3][j & 7];
  float g[8];
#pragma unroll
  for (int k = 0; k < 8; ++k) {
    float s = 0.f;
#pragma unroll
    for (int n = 0; n < 8; ++n)
      if ((j >> (7 - n)) & 1) s += bas[n][k];  // grid[j][n] = (j >> (7-n)) & 1
    g[k] = s;
  }
  float gg = 0.f;
#pragma unroll
  for (int k = 0; k < 8; ++k) { ws[64 + j * 8 + k] = g[k]; gg += g[k] * g[k]; }
  ws[64 + 2048 + j] = gg;
}

__global__ __launch_bounds__(256) void lvq_main_kernel(
    const float* __restrict__ x, const float* __restrict__ ws,
    float* __restrict__ out, int B) {
  const float* basis = ws;
  const float* G = ws + 64;
  const float* gg = ws + 64 + 2048;
  float* out_xh   = out;
  float* out_xh2  = out + (size_t)B * 8;
  float* out_dist = out + (size_t)B * 16;

  const int lane = threadIdx.x & 31;
  const int wid  = threadIdx.x >> 5;
  const int wave = blockIdx.x * (blockDim.x >> 5) + wid;
  const long base_row = (long)wave * 16;
  if (base_row >= B) return;
  const int  m  = lane & 15;
  const bool hi = lane >= 16;

  // each lane owns batch row (base_row + lane%16); halves are duplicates,
  // which is exactly the A-operand layout for V_WMMA_F32_16X16X4_F32.
  const float* xr = x + (base_row + m) * 8;
  float xo[8], xt[8];
#pragma unroll
  for (int k = 0; k < 8; ++k) { xo[k] = xr[k]; xt[k] = xo[k]; }

  // triangular (Babai) reduction
  float us[8];
#pragma unroll
  for (int i = 7; i >= 0; --i) {
    float ui = floorf(xt[i] / basis[i * 8 + i]);
    us[i] = ui;
#pragma unroll
    for (int k = 0; k < 8; ++k)
      if (k < i) xt[k] -= ui * basis[i * 8 + k];
  }
  // c0 = dot @ basis  (basis lower-triangular)
  float c0[8];
#pragma unroll
  for (int k = 0; k < 8; ++k) {
    float s = 0.f;
#pragma unroll
    for (int n = 0; n < 8; ++n)
      if (n >= k) s += us[n] * basis[n * 8 + k];
    c0[k] = s;
  }
  // v = c0 - x ; dist_j = (||v||^2 + 2 v.G_j + ||G_j||^2)/8
  float v[8], vv = 0.f;
#pragma unroll
  for (int k = 0; k < 8; ++k) { v[k] = c0[k] - xo[k]; vv += v[k] * v[k]; }

  // A operands: 16x4 f32 tiles (K=0..3 and K=4..7)
  v2f a0, a1;
  a0.x = hi ? v[2] : v[0];  a0.y = hi ? v[3] : v[1];
  a1.x = hi ? v[6] : v[4];  a1.y = hi ? v[7] : v[5];

  float best[8]; int bidx[8];
#pragma unroll
  for (int r = 0; r < 8; ++r) { best[r] = 3.4e38f; bidx[r] = 0; }

#pragma unroll
  for (int t = 0; t < 16; ++t) {           // 16 codeword tiles of 16
    const float* gr = G + (t * 16 + m) * 8;
    float g[8];
#pragma unroll
    for (int k = 0; k < 8; ++k) g[k] = gr[k];
    // B operands: 4x16 f32, column n = lane%16, rows K per half-wave
    v2f b0, b1;
    b0.x = hi ? g[2] : g[0];  b0.y = hi ? g[3] : g[1];
    b1.x = hi ? g[6] : g[4];  b1.y = hi ? g[7] : g[5];
    float ggn = gg[t * 16 + m];

    v8f acc = {};
    acc = __builtin_amdgcn_wmma_f32_16x16x4_f32(false, a0, false, b0,
                                                (short)0, acc, false, false);
    acc = __builtin_amdgcn_wmma_f32_16x16x4_f32(false, a1, false, b1,
                                                (short)0, acc, false, false);
#pragma unroll
    for (int r = 0; r < 8; ++r) {
      int Mrow = r + (hi ? 8 : 0);                  // C/D layout rows
      float vvr = __shfl(vv, Mrow, 32);
      float d = (vvr + 2.f * acc[r] + ggn) * 0.125f;
      // pure streaming store, never re-read: non-temporal hint keeps it
      // from dirtying L2 (dist alone is ~134 MB of write-once data)
      __builtin_nontemporal_store(
          d, &out_dist[(size_t)(base_row + Mrow) * 256 + t * 16 + m]);
      int j = t * 16 + m;
      if (d < best[r] || (d == best[r] && j < bidx[r])) { best[r] = d; bidx[r] = j; }
    }
  }

  // argmin across the 16 lanes of each half-wave (xor stays within halves)
#pragma unroll
  for (int off = 1; off < 16; off <<= 1) {
#pragma unroll
    for (int r = 0; r < 8; ++r) {
      float ov = __shfl_xor(best[r], off, 32);
      int   oi = __shfl_xor(bidx[r], off, 32);
      if (ov < best[r] || (ov == best[r] && oi < bidx[r])) { best[r] = ov; bidx[r] = oi; }
    }
  }

  __shared__ int sidx[8][16];  // per-wave row -> argmin index
  if (lane == 0) {
#pragma unroll
    for (int r = 0; r < 8; ++r) sidx[wid][r] = bidx[r];
  }
  if (lane == 16) {
#pragma unroll
    for (int r = 0; r < 8; ++r) sidx[wid][8 + r] = bidx[r];
  }
  __syncthreads();

  if (!hi) {  // lane m emits x_hat for its own row: c0 + G[idx]
    int idx = sidx[wid][m];
    const float* gi = G + idx * 8;
#pragma unroll
    for (int k = 0; k < 8; ++k) {
      float xh = c0[k] + gi[k];
      size_t o = (size_t)(base_row + m) * 8 + k;
      __builtin_nontemporal_store(xh, &out_xh[o]);
      __builtin_nontemporal_store(xh, &out_xh2[o]);
    }
  }
}

extern "C" void kernel_launch(void* const* d_in, const int* in_sizes, int n_in,
                              void* d_out, int out_size, void* d_ws, size_t ws_size,
                              hipStream_t stream) {
  const float* x  = (const float*)d_in[0];   // [B,8] f32
  const float* bp = (const float*)d_in[1];   // [8,8] f32 basis_param
  (void)n_in; (void)out_size; (void)ws_size;
  float* out = (float*)d_out;                // [B*8 | B*8 | B*256] f32
  float* ws  = (float*)d_ws;
  int B = in_sizes[0] / 8;

  lvq_prep_kernel<<<1, 256, 0, stream>>>(bp, ws);
  int waves  = (B + 15) / 16;
  int blocks = (waves + 7) / 8;              // 8 waves (256 threads) per block
  lvq_main_kernel<<<blocks, 256, 0, stream>>>(x, ws, out, B);
}